// TinyBird_60936995996235
// MI455X (gfx1250) — compile-verified
//
#include <hip/hip_runtime.h>
#include <hip/hip_bf16.h>
#include <stdint.h>

#define B_    16
#define H_    128
#define W_    8192
#define NMAX  128
#define WOUT  6271
#define SEP_ID 2
#define BIGVAL (1 << 30)
#define TILE_T 256
#define ROWS_Z 16   // H_ / gridDim.z

#define GLOBAL_AS __attribute__((address_space(1)))
#define LDS_AS    __attribute__((address_space(3)))

// ---------------------------------------------------------------------------
// Kernel 3 (emitted first so its asm shows in the snippet): the bandwidth
// kernel. Each block: one 256-column tile of one batch, 16 H-rows (grid.z
// splits H). Plan tile staged in LDS once via CDNA5 async global->LDS copy
// (global_load_async_to_lds_b32 + s_wait_asynccnt), reused by all 16 rows.
// Stores are fully coalesced along t; gather reads hit L2 (x fits in 192 MB).
// ---------------------------------------------------------------------------
__global__ __launch_bounds__(TILE_T) void k_gather(
    const float* __restrict__ x, const float* __restrict__ sep_p,
    const int* __restrict__ plan, float* __restrict__ x_new)
{
    const int b   = blockIdx.y;
    const int t0  = blockIdx.x * TILE_T;
    const int h0  = blockIdx.z * ROWS_Z;
    const int tid = threadIdx.x;

    __shared__ int plan_s[TILE_T];
    const int nvalid = (WOUT - t0) < TILE_T ? (WOUT - t0) : TILE_T;
    const int* gplan = plan + (size_t)b * WOUT + t0;

#if defined(__gfx1250__) && __has_builtin(__builtin_amdgcn_global_load_async_to_lds_b32) && __has_builtin(__builtin_amdgcn_s_wait_asynccnt)
    if (tid < nvalid) {
        __builtin_amdgcn_global_load_async_to_lds_b32(
            (GLOBAL_AS int*)(uintptr_t)(gplan + tid),
            (LDS_AS int*)(&plan_s[tid]),
            0, 0);
    }
    __builtin_amdgcn_s_wait_asynccnt(0);   // each wave drains its own ASYNCcnt
#else
    if (tid < nvalid) plan_s[tid] = gplan[tid];
#endif
    __syncthreads();                       // cross-wave visibility of LDS tile

    if (tid >= nvalid) return;             // no later barriers; tail lanes out

    const float sep  = sep_p[0];
    const int   code = plan_s[tid];
    const int   t    = t0 + tid;

    const float* xrow = x     + ((size_t)b * H_ + h0) * (size_t)W_;
    float*       orow = x_new + ((size_t)b * H_ + h0) * (size_t)WOUT;

    #pragma unroll
    for (int h = 0; h < ROWS_Z; ++h) {
        float v;
        if (code >= 0)       v = xrow[code];   // gather (L2-resident)
        else if (code == -1) v = sep;
        else                 v = 0.0f;
        orow[t] = v;                           // coalesced 128B/wave store
        xrow += W_;
        orow += WOUT;
    }
}

// ---------------------------------------------------------------------------
// Kernel 1: per-batch prefix scan over chirp widths -> sn/en, write xi_new.
// One block of 128 threads per batch (tiny).
// ---------------------------------------------------------------------------
__global__ __launch_bounds__(NMAX) void k_scan(
    const int* __restrict__ xi, const int* __restrict__ N,
    int* __restrict__ sn_g, int* __restrict__ en_g, int* __restrict__ xi_new)
{
    const int b = blockIdx.x;
    const int j = threadIdx.x;
    __shared__ int s[NMAX];

    const int Nb = N[b];
    const int s0 = xi[(b * NMAX + j) * 2 + 0];
    const int e0 = xi[(b * NMAX + j) * 2 + 1];
    int w = e0 - s0; if (w < 0) w = 0;
    const bool valid = (j < Nb);
    const int wv = valid ? w : 0;

    s[j] = wv;
    __syncthreads();
    // Hillis-Steele inclusive scan over 128 entries
    for (int off = 1; off < NMAX; off <<= 1) {
        const int v = (j >= off) ? s[j - off] : 0;
        __syncthreads();
        s[j] += v;
        __syncthreads();
    }
    const int csum = s[j];
    const int sn = csum - wv + j;   // seps before = j
    const int en = csum + j;

    sn_g[b * NMAX + j] = sn;
    en_g[b * NMAX + j] = en;
    xi_new[(b * NMAX + j) * 2 + 0] = valid ? sn : 0;
    xi_new[(b * NMAX + j) * 2 + 1] = valid ? en : 0;
}

// ---------------------------------------------------------------------------
// Kernel 2: build per-(b,t) plan code and xl_new.
//   code >= 0  -> in_chirp, code = src column in x
//   code == -1 -> separator column
//   code == -2 -> zero fill
// Binary search over en (strictly increasing on valid prefix) held in LDS.
// ---------------------------------------------------------------------------
__global__ __launch_bounds__(TILE_T) void k_plan(
    const int* __restrict__ xi, const int* __restrict__ N,
    const int* __restrict__ xl,
    const int* __restrict__ sn_g, const int* __restrict__ en_g,
    int* __restrict__ plan, int* __restrict__ xl_new)
{
    const int b   = blockIdx.y;
    const int t   = blockIdx.x * TILE_T + threadIdx.x;
    const int tid = threadIdx.x;

    __shared__ int sn_s[NMAX], en_s[NMAX], s0_s[NMAX], e0_s[NMAX];
    if (tid < NMAX) {
        sn_s[tid] = sn_g[b * NMAX + tid];
        en_s[tid] = en_g[b * NMAX + tid];
        s0_s[tid] = xi[(b * NMAX + tid) * 2 + 0];
        e0_s[tid] = xi[(b * NMAX + tid) * 2 + 1];
    }
    __syncthreads();
    if (t >= WOUT) return;

    const int Nb = N[b];
    // i = count of (valid en <= t): lower_bound over en_srt (invalid -> BIG)
    int lo = 0, hi = NMAX;
    while (lo < hi) {
        const int mid = (lo + hi) >> 1;
        const int e = (mid < Nb) ? en_s[mid] : BIGVAL;
        if (e <= t) lo = mid + 1; else hi = mid;
    }
    const int i  = lo;
    const int ic = (i < NMAX - 1) ? i : (NMAX - 1);

    const bool in_chirp = (i < Nb) && (t >= sn_s[ic]) &&
                          (en_s[ic] > sn_s[ic]) && (e0_s[ic] > s0_s[ic]);
    int src = s0_s[ic] + (t - sn_s[ic]);
    src = src < 0 ? 0 : (src > W_ - 1 ? W_ - 1 : src);

    int code;
    if (in_chirp) {
        code = src;
    } else {
        int jp = i - 1;
        jp = jp < 0 ? 0 : (jp > NMAX - 1 ? NMAX - 1 : jp);
        const int en_prev = en_s[jp];
        const bool is_sep = (i >= 1) && (t == en_prev) &&
                            ((i - 1) < (Nb - 1)) && (en_prev < W_);
        code = is_sep ? -1 : -2;
    }
    plan[(size_t)b * WOUT + t] = code;
    xl_new[(size_t)b * WOUT + t] =
        (code >= 0) ? xl[(size_t)b * W_ + code] : (code == -1 ? SEP_ID : 0);
}

// ---------------------------------------------------------------------------
extern "C" void kernel_launch(void* const* d_in, const int* in_sizes, int n_in,
                              void* d_out, int out_size, void* d_ws, size_t ws_size,
                              hipStream_t stream) {
    (void)in_sizes; (void)n_in; (void)out_size; (void)ws_size;

    const float* x   = (const float*)d_in[0];
    const int*   xi  = (const int*)  d_in[1];
    const int*   N   = (const int*)  d_in[2];
    const int*   xl  = (const int*)  d_in[3];
    const float* sep = (const float*)d_in[4];
    // d_in[5] (w_out) is a compile-time constant of the reference: WOUT = 6271

    // Output layout (flat, return order): x_new f32 | xi_new i32 | xl_new i32
    float* x_new  = (float*)d_out;
    int*   xi_new = (int*)((float*)d_out + (size_t)B_ * H_ * WOUT);
    int*   xl_new = xi_new + (size_t)B_ * NMAX * 2;

    // Workspace: sn | en | plan
    int* sn_g = (int*)d_ws;
    int* en_g = sn_g + (size_t)B_ * NMAX;
    int* plan = en_g + (size_t)B_ * NMAX;

    const int tiles_t = (WOUT + TILE_T - 1) / TILE_T;

    k_scan<<<dim3(B_), dim3(NMAX), 0, stream>>>(xi, N, sn_g, en_g, xi_new);
    k_plan<<<dim3(tiles_t, B_), dim3(TILE_T), 0, stream>>>(
        xi, N, xl, sn_g, en_g, plan, xl_new);
    k_gather<<<dim3(tiles_t, B_, H_ / ROWS_Z), dim3(TILE_T), 0, stream>>>(
        x, sep, plan, x_new);
}